// GAT_9363028706300
// MI455X (gfx1250) — compile-verified
//
#include <hip/hip_runtime.h>
#include <math.h>

typedef __attribute__((ext_vector_type(2))) float v2f;
typedef __attribute__((ext_vector_type(8))) float v8f;

#define LRELU 0.2f
#define LN_EPS 1e-5f

// ---------------------------------------------------------------------------
// GEMM: Hout[N, DOUT] = X[N, K] @ W[K, DOUT], fp32 via V_WMMA_F32_16X16X4_F32.
// One wave per 16x16 output tile. Layouts per CDNA5 ISA 7.12.2:
//   A 16x4 : lanes 0-15 -> M=lane, {v0,v1}={K0,K1}; lanes 16-31 -> {K2,K3}
//   B 4x16 : lanes 0-15 -> N=lane, {v0,v1}={K0,K1}; lanes 16-31 -> {K2,K3}
//   C/D    : vgpr i -> M = i + 8*(lane>=16), N = lane&15
// ---------------------------------------------------------------------------
__global__ __launch_bounds__(32)
void gemm_wmma_f32(const float* __restrict__ X, const float* __restrict__ W,
                   float* __restrict__ Hout, int K, int DOUT) {
    const int lane = threadIdx.x & 31;
    const int half = lane >> 4;        // 0: K0/K1,  1: K2/K3
    const int lr   = lane & 15;
    const int n0   = blockIdx.x * 16;
    const int m0   = blockIdx.y * 16;

    const float* xrow = X + (size_t)(m0 + lr) * K;
    v8f c = {};
    for (int k = 0; k < K; k += 4) {
        const int ka = k + 2 * half;
        v2f a, b;
        a.x = xrow[ka];
        a.y = xrow[ka + 1];
        b.x = W[(size_t)ka * DOUT + n0 + lr];
        b.y = W[(size_t)(ka + 1) * DOUT + n0 + lr];
        c = __builtin_amdgcn_wmma_f32_16x16x4_f32(false, a, false, b,
                                                  (short)0, c, false, false);
    }
    #pragma unroll
    for (int i = 0; i < 8; ++i)
        Hout[(size_t)(m0 + 8 * half + i) * DOUT + n0 + lr] = c[i];
}

// ---------------------------------------------------------------------------
// als[n,h] = sum_c H[n,h,c]*a_src[h,c] ; ald likewise. One thread / (node,head)
// ---------------------------------------------------------------------------
__global__ void attn_coef(const float* __restrict__ H, const float* __restrict__ aS,
                          const float* __restrict__ aD, float* __restrict__ als,
                          float* __restrict__ ald, int N_, int heads, int ch) {
    int gid = blockIdx.x * blockDim.x + threadIdx.x;
    if (gid >= N_ * heads) return;
    const int node = gid / heads, hd = gid % heads;
    const float* hp = H + (size_t)node * heads * ch + (size_t)hd * ch;
    const float* sp = aS + hd * ch;
    const float* dp = aD + hd * ch;
    float s = 0.f, d = 0.f;
    for (int c = 0; c < ch; ++c) { float v = hp[c]; s += v * sp[c]; d += v * dp[c]; }
    als[gid] = s;
    ald[gid] = d;
}

__global__ void init_bufs(float* __restrict__ emax, float* __restrict__ denom,
                          float* __restrict__ agg, int nh, int nfeat) {
    int gid = blockIdx.x * blockDim.x + threadIdx.x;
    if (gid < nh)    { emax[gid] = -3.0e38f; denom[gid] = 0.f; }
    if (gid < nfeat) { agg[gid] = 0.f; }
}

__device__ inline void atomicMaxF32(float* addr, float val) {
    unsigned int* ua = (unsigned int*)addr;
    unsigned int old = *ua;
    while (__uint_as_float(old) < val) {
        unsigned int assumed = old;
        old = atomicCAS(ua, assumed, __float_as_uint(val));
        if (old == assumed) break;
    }
}

__device__ inline void edge_ends(int eid, int E_, const int* __restrict__ src,
                                 const int* __restrict__ dst, int& s, int& d) {
    if (eid < E_) { s = src[eid]; d = dst[eid]; }
    else          { s = d = eid - E_; }          // self-loops appended
}

__device__ inline float edge_logit(const float* als, const float* ald,
                                   int s, int d, int heads, int hd) {
    float e = als[s * heads + hd] + ald[d * heads + hd];
    return e > 0.f ? e : LRELU * e;
}

__global__ void edge_max_k(const int* __restrict__ src, const int* __restrict__ dst,
                           const float* __restrict__ als, const float* __restrict__ ald,
                           float* __restrict__ emax, int E_, int N_, int heads) {
    int gid = blockIdx.x * blockDim.x + threadIdx.x;
    if (gid >= (E_ + N_) * heads) return;
    int s, d; edge_ends(gid / heads, E_, src, dst, s, d);
    const int hd = gid % heads;
    atomicMaxF32(&emax[d * heads + hd], edge_logit(als, ald, s, d, heads, hd));
}

__global__ void edge_sum_k(const int* __restrict__ src, const int* __restrict__ dst,
                           const float* __restrict__ als, const float* __restrict__ ald,
                           const float* __restrict__ emax, float* __restrict__ denom,
                           int E_, int N_, int heads) {
    int gid = blockIdx.x * blockDim.x + threadIdx.x;
    if (gid >= (E_ + N_) * heads) return;
    int s, d; edge_ends(gid / heads, E_, src, dst, s, d);
    const int hd = gid % heads;
    const int dh = d * heads + hd;
    atomicAdd(&denom[dh], __expf(edge_logit(als, ald, s, d, heads, hd) - emax[dh]));
}

__global__ void edge_aggr_k(const int* __restrict__ src, const int* __restrict__ dst,
                            const float* __restrict__ als, const float* __restrict__ ald,
                            const float* __restrict__ emax, const float* __restrict__ denom,
                            const float* __restrict__ H, float* __restrict__ agg,
                            int E_, int N_, int heads, int ch) {
    int gid = blockIdx.x * blockDim.x + threadIdx.x;
    if (gid >= (E_ + N_) * heads) return;
    int s, d; edge_ends(gid / heads, E_, src, dst, s, d);
    const int hd = gid % heads;
    const int dh = d * heads + hd;
    const float alpha = __expf(edge_logit(als, ald, s, d, heads, hd) - emax[dh]) / denom[dh];
    const int ldh = heads * ch;
    const float4* hp = (const float4*)(H + (size_t)s * ldh + (size_t)hd * ch);
    float* op = agg + (size_t)d * ldh + (size_t)hd * ch;
    for (int i = 0; i < (ch >> 2); ++i) {
        float4 v = hp[i];
        atomicAdd(op + 4 * i + 0, v.x * alpha);
        atomicAdd(op + 4 * i + 1, v.y * alpha);
        atomicAdd(op + 4 * i + 2, v.z * alpha);
        atomicAdd(op + 4 * i + 3, v.w * alpha);
    }
}

// ---------------------------------------------------------------------------
// out = relu(layernorm(agg + bias))   — one wave32 per node, D/32 vals per lane
// ---------------------------------------------------------------------------
__global__ void bias_ln_relu(const float* __restrict__ agg, const float* __restrict__ bias,
                             const float* __restrict__ g, const float* __restrict__ b,
                             float* __restrict__ out, int N_, int D) {
    const int lane = threadIdx.x & 31;
    const int node = blockIdx.x * (blockDim.x >> 5) + (threadIdx.x >> 5);
    if (node >= N_) return;
    const int per = D >> 5;                     // 4 for D=128
    const float* xp = agg + (size_t)node * D;
    float vals[8];
    float s = 0.f;
    for (int i = 0; i < per; ++i) { vals[i] = xp[lane * per + i] + bias[lane * per + i]; s += vals[i]; }
    for (int o = 16; o > 0; o >>= 1) s += __shfl_xor(s, o, 32);
    const float mu = s / (float)D;
    float q = 0.f;
    for (int i = 0; i < per; ++i) { float t = vals[i] - mu; q += t * t; }
    for (int o = 16; o > 0; o >>= 1) q += __shfl_xor(q, o, 32);
    const float inv = rsqrtf(q / (float)D + LN_EPS);
    for (int i = 0; i < per; ++i) {
        const int c = lane * per + i;
        float y = (vals[i] - mu) * inv * g[c] + b[c];
        out[(size_t)node * D + c] = y > 0.f ? y : 0.f;
    }
}

// ---------------------------------------------------------------------------
// out = log_softmax(agg + bias)       — one wave32 per node (D=64, 2 per lane)
// ---------------------------------------------------------------------------
__global__ void bias_logsoftmax(const float* __restrict__ agg, const float* __restrict__ bias,
                                float* __restrict__ out, int N_, int D) {
    const int lane = threadIdx.x & 31;
    const int node = blockIdx.x * (blockDim.x >> 5) + (threadIdx.x >> 5);
    if (node >= N_) return;
    const int per = D >> 5;                     // 2 for D=64
    const float* xp = agg + (size_t)node * D;
    float vals[8];
    float mx = -3.0e38f;
    for (int i = 0; i < per; ++i) { vals[i] = xp[lane * per + i] + bias[lane * per + i]; mx = fmaxf(mx, vals[i]); }
    for (int o = 16; o > 0; o >>= 1) mx = fmaxf(mx, __shfl_xor(mx, o, 32));
    float s = 0.f;
    for (int i = 0; i < per; ++i) s += __expf(vals[i] - mx);
    for (int o = 16; o > 0; o >>= 1) s += __shfl_xor(s, o, 32);
    const float lse = mx + logf(s);
    for (int i = 0; i < per; ++i)
        out[(size_t)node * D + lane * per + i] = vals[i] - lse;
}

// ---------------------------------------------------------------------------
extern "C" void kernel_launch(void* const* d_in, const int* in_sizes, int n_in,
                              void* d_out, int out_size, void* d_ws, size_t ws_size,
                              hipStream_t stream) {
    const float* x    = (const float*)d_in[0];
    const int*   eidx = (const int*)  d_in[1];
    const float* W0   = (const float*)d_in[2];
    const float* as0  = (const float*)d_in[3];
    const float* ad0  = (const float*)d_in[4];
    const float* b0   = (const float*)d_in[5];
    const float* W1   = (const float*)d_in[6];
    const float* as1  = (const float*)d_in[7];
    const float* ad1  = (const float*)d_in[8];
    const float* b1   = (const float*)d_in[9];
    const float* W2   = (const float*)d_in[10];
    const float* as2  = (const float*)d_in[11];
    const float* ad2  = (const float*)d_in[12];
    const float* b2   = (const float*)d_in[13];
    const float* ln1g = (const float*)d_in[14];
    const float* ln1b = (const float*)d_in[15];
    const float* ln2g = (const float*)d_in[16];
    const float* ln2b = (const float*)d_in[17];

    const int N_ = in_sizes[0] / 128;   // 100000
    const int E_ = in_sizes[1] / 2;     // 1600000
    const int* src = eidx;
    const int* dst = eidx + E_;

    float* ws    = (float*)d_ws;
    float* hbuf  = ws;                          // N*128  (layer input)
    float* gbuf  = hbuf  + (size_t)N_ * 128;    // N*128  (post-GEMM h)
    float* agg   = gbuf  + (size_t)N_ * 128;    // N*128  (aggregated)
    float* als   = agg   + (size_t)N_ * 128;    // N*8
    float* ald   = als   + (size_t)N_ * 8;      // N*8
    float* emax  = ald   + (size_t)N_ * 8;      // N*8
    float* denom = emax  + (size_t)N_ * 8;      // N*8

    auto gat_layer = [&](const float* X, const float* W, const float* aS,
                         const float* aD, int heads, int ch) {
        const int dout = heads * ch;
        dim3 gg(dout / 16, N_ / 16);
        gemm_wmma_f32<<<gg, 32, 0, stream>>>(X, W, gbuf, 128, dout);

        const int nh = N_ * heads;
        attn_coef<<<(nh + 255) / 256, 256, 0, stream>>>(gbuf, aS, aD, als, ald, N_, heads, ch);

        const int nfeat = N_ * dout;
        const int ninit = nfeat > nh ? nfeat : nh;
        init_bufs<<<(ninit + 255) / 256, 256, 0, stream>>>(emax, denom, agg, nh, nfeat);

        const int tot = (E_ + N_) * heads;
        const int blocks = (tot + 255) / 256;
        edge_max_k <<<blocks, 256, 0, stream>>>(src, dst, als, ald, emax, E_, N_, heads);
        edge_sum_k <<<blocks, 256, 0, stream>>>(src, dst, als, ald, emax, denom, E_, N_, heads);
        edge_aggr_k<<<blocks, 256, 0, stream>>>(src, dst, als, ald, emax, denom, gbuf, agg,
                                                E_, N_, heads, ch);
    };

    const int ln_blocks = (N_ + 7) / 8;         // 8 wave32's per 256-thread block

    // Layer 0: 128 -> 8 heads x 16
    gat_layer(x, W0, as0, ad0, 8, 16);
    bias_ln_relu<<<ln_blocks, 256, 0, stream>>>(agg, b0, ln1g, ln1b, hbuf, N_, 128);

    // Layer 1: 128 -> 8 heads x 16
    gat_layer(hbuf, W1, as1, ad1, 8, 16);
    bias_ln_relu<<<ln_blocks, 256, 0, stream>>>(agg, b1, ln2g, ln2b, hbuf, N_, 128);

    // Layer 2: 128 -> 1 head x 64, then log_softmax
    gat_layer(hbuf, W2, as2, ad2, 1, 64);
    bias_logsoftmax<<<ln_blocks, 256, 0, stream>>>(agg, b2, (float*)d_out, N_, 64);
}